// DiscourseParser_28587302322807
// MI455X (gfx1250) — compile-verified
//
#include <hip/hip_runtime.h>
#include <math.h>

// Problem constants (DiscourseParser reference): B=4, S=256, H=768, R=7
#define BDIM 4
#define SDIM 256
#define HDIM 768
#define RDIM 7
#define BS   (BDIM * SDIM)          // 1024 rows for the two projections
#define SPAD 772                    // LDS row stride: 772 % 64 == 4 -> conflict-free b64 reads
#define TCH  8                      // t-chunks per (b, s-tile); 32 t's per block, 4 per wave

typedef __attribute__((ext_vector_type(2))) float v2f;
typedef __attribute__((ext_vector_type(8))) float v8f;

// Fast hardware tanh (V_TANH_F32, TRANS32-class on gfx1250) with safe fallback.
#if __has_builtin(__builtin_amdgcn_tanhf)
__device__ __forceinline__ float fast_tanh(float x) { return __builtin_amdgcn_tanhf(x); }
#elif __has_builtin(__builtin_amdgcn_tanh_f32)
__device__ __forceinline__ float fast_tanh(float x) { return __builtin_amdgcn_tanh_f32(x); }
#else
__device__ __forceinline__ float fast_tanh(float x) { return tanhf(x); }
#endif

// ---------------------------------------------------------------------------
// Stage 0: build zero-padded w_out^T table in workspace.
// Entry (h, n) lives at wpad[(h>>1)*32 + n*2 + (h&1)] so that one 8-byte load
// per lane fetches the V_WMMA_F32_16X16X4_F32 B-fragment (K=h0..h0+3, N=n).
// n in [7,16) is zero (R=7 padded to the 16-wide WMMA tile).
// ---------------------------------------------------------------------------
__global__ __launch_bounds__(256)
void build_wpad_kernel(const float* __restrict__ wout, float* __restrict__ wpad) {
    int i = blockIdx.x * 256 + threadIdx.x;          // over H*16 = 12288 entries
    if (i < HDIM * 16) {
        int h = i >> 4, n = i & 15;
        wpad[(h >> 1) * 32 + n * 2 + (h & 1)] = (n < RDIM) ? wout[n * HDIM + h] : 0.0f;
    }
}

// ---------------------------------------------------------------------------
// Stage 1: Y[m,k] = sum_h X[m,h] * W[k,h] + bias[k]    (X:[BS,H], W:[H,H])
// One 16x16 output tile per wave, K-loop of H/4 V_WMMA_F32_16X16X4_F32 ops.
// A-frag layout: lane l -> row M = l%16, K = 2*(l/16) + {0,1}  (float2 load)
// B-frag layout: lane l -> col N = l%16, K = 2*(l/16) + {0,1}  (W row-major)
// C/D layout:    VGPR j, lane l -> D[j + 8*(l/16)][l%16]
// ---------------------------------------------------------------------------
__global__ __launch_bounds__(256)
void linear_wmma_kernel(const float* __restrict__ X, const float* __restrict__ W,
                        const float* __restrict__ bias, float* __restrict__ Y) {
    const int lane = threadIdx.x & 31;
    const int wave = threadIdx.x >> 5;               // 0..7
    const int m0   = blockIdx.x * 16;                // M tile (64 tiles)
    const int n0   = blockIdx.y * 128 + wave * 16;   // N tile (6*8 = 48 tiles)
    const int row  = lane & 15;
    const int kh   = (lane >> 4) * 2;                // 0 or 2

    const float* xrow = X + (m0 + row) * HDIM + kh;
    const float* wrow = W + (n0 + row) * HDIM + kh;

    v8f c;
    float bv = bias[n0 + row];                       // C[m][n] = bias[n] for all m
    #pragma unroll
    for (int j = 0; j < 8; ++j) c[j] = bv;

    #pragma unroll 4
    for (int h0 = 0; h0 < HDIM; h0 += 4) {
        v2f a = *(const v2f*)(xrow + h0);
        v2f b = *(const v2f*)(wrow + h0);
        c = __builtin_amdgcn_wmma_f32_16x16x4_f32(false, a, false, b,
                                                  (short)0, c, false, false);
    }

    float* yout = Y + (m0 + (lane >> 4) * 8) * HDIM + n0 + row;
    #pragma unroll
    for (int j = 0; j < 8; ++j) yout[j * HDIM] = c[j];
}

// ---------------------------------------------------------------------------
// Stage 2 (dominant): out[b,s,t,r] = sum_h tanh(h_src[b,s,h]+h_tgt[b,t,h]) * w_out[r,h]
// Block: fixed b, one 16-row s-tile (staged in LDS, padded stride), 32 t's.
// 8 waves x 4 t's each; per t a 16(M=s) x 16(N=r, 7 real) x 768(K) fp32 WMMA GEMM.
// A-frag built on the fly: LDS b64 (h_src) + global b64 (h_tgt, broadcast row,
// L0-resident) + 2 adds + 2 v_tanh_f32 per lane per K-step. WMMA (fp32, XDL)
// overlaps with the TRANS/VALU work of the next fragment.
// ---------------------------------------------------------------------------
__global__ __launch_bounds__(256)
void pair_tanh_wmma_kernel(const float* __restrict__ HS, const float* __restrict__ HT,
                           const float* __restrict__ WPAD, float* __restrict__ OUT) {
    __shared__ float lds_hs[16 * SPAD];              // ~49.4 KB

    const int tid  = threadIdx.x;
    const int lane = tid & 31;
    const int wave = tid >> 5;

    const int bs_tile = blockIdx.x;                  // 0..63 = b*16 + s_tile
    const int b  = bs_tile >> 4;
    const int s0 = (bs_tile & 15) * 16;
    const int t0 = blockIdx.y * (SDIM / TCH);        // 32-wide t chunk

    // cooperative load of the 16 x H h_src tile into padded LDS
    for (int i = tid; i < 16 * HDIM; i += 256) {
        int r_ = i / HDIM, cc = i - r_ * HDIM;
        lds_hs[r_ * SPAD + cc] = HS[(b * SDIM + s0 + r_) * HDIM + cc];
    }
    __syncthreads();

    const int row = lane & 15;                       // M row (s) / N col (r)
    const int kh  = (lane >> 4) * 2;                 // K sub-offset 0 or 2

    const float* hsp = &lds_hs[row * SPAD + kh];
    const float* wp  = WPAD + (kh >> 1) * 32 + row * 2;   // advances 64 floats / K-step

    for (int tt = 0; tt < (SDIM / TCH); tt += 8) {
        const int t = t0 + tt + wave;
        const float* htp = HT + (b * SDIM + t) * HDIM + kh;

        v8f c = {};
        #pragma unroll 4
        for (int i = 0; i < HDIM / 4; ++i) {
            v2f hs = *(const v2f*)(hsp + i * 4);
            v2f ht = *(const v2f*)(htp + i * 4);
            v2f a;
            a.x = fast_tanh(hs.x + ht.x);
            a.y = fast_tanh(hs.y + ht.y);
            v2f bf = *(const v2f*)(wp + i * 64);
            c = __builtin_amdgcn_wmma_f32_16x16x4_f32(false, a, false, bf,
                                                      (short)0, c, false, false);
        }
        // D[s_local = j + 8*(lane>>4)][r = row]; keep r < 7 only.
        if (row < RDIM) {
            float* o = OUT + ((size_t)(b * SDIM + s0 + (lane >> 4) * 8) * SDIM + t) * RDIM + row;
            #pragma unroll
            for (int j = 0; j < 8; ++j) o[(size_t)j * (SDIM * RDIM)] = c[j];
        }
    }
}

// ---------------------------------------------------------------------------
extern "C" void kernel_launch(void* const* d_in, const int* in_sizes, int n_in,
                              void* d_out, int out_size, void* d_ws, size_t ws_size,
                              hipStream_t stream) {
    const float* pooled = (const float*)d_in[0];     // [4,256,768]
    const float* w_src  = (const float*)d_in[1];     // [768,768]
    const float* b_src  = (const float*)d_in[2];     // [768]
    const float* w_tgt  = (const float*)d_in[3];     // [768,768]
    const float* b_tgt  = (const float*)d_in[4];     // [768]
    const float* w_out  = (const float*)d_in[5];     // [7,768]
    float* out = (float*)d_out;                      // [4,256,256,7]

    // workspace layout (floats): h_src [1024*768] | h_tgt [1024*768] | wpad [12288]
    float* hs   = (float*)d_ws;
    float* ht   = hs + (size_t)BS * HDIM;
    float* wpad = ht + (size_t)BS * HDIM;            // total ~6.4 MB

    build_wpad_kernel<<<dim3((HDIM * 16 + 255) / 256), 256, 0, stream>>>(w_out, wpad);

    dim3 g1(BS / 16, HDIM / 128);                    // 64 x 6 blocks, 8 waves each
    linear_wmma_kernel<<<g1, 256, 0, stream>>>(pooled, w_src, b_src, hs);
    linear_wmma_kernel<<<g1, 256, 0, stream>>>(pooled, w_tgt, b_tgt, ht);

    dim3 g2(BDIM * (SDIM / 16), TCH);                // 64 x 8 blocks, 4096 waves
    pair_tanh_wmma_kernel<<<g2, 256, 0, stream>>>(hs, ht, wpad, out);
}